// Luong2_65807488909368
// MI455X (gfx1250) — compile-verified
//
#include <hip/hip_runtime.h>
#include <hip/hip_bf16.h>
#include <cmath>

typedef __attribute__((ext_vector_type(2))) float v2f;
typedef __attribute__((ext_vector_type(8))) float v8f;

#define B_ 16
#define T_ 8192
#define D_ 512
#define U_ 128

// ---------------------------------------------------------------------------
// Kernel 1: score[b,t] = (values[b,t,:] . query[b,:]) * sum(W) + sum(bias)
// via V_WMMA_F32_16X16X4_F32. One wave handles a 16-row t-tile; query chunk is
// replicated across the 16 B-matrix columns, so every output column equals the
// dot products; lanes with N==0 (lane 0 / lane 16) extract rows 0-7 / 8-15.
// A-matrix 16x4 f32 layout (ISA 7.12.2): lane L holds M=L&15,
// K = (L>=16 ? 2 : 0)+v in VGPR v. Replicated B uses the same K mapping.
// ---------------------------------------------------------------------------
__global__ __launch_bounds__(128) void scores_wmma(
    const float* __restrict__ query,
    const float* __restrict__ values,
    const float* __restrict__ W,
    const float* __restrict__ bias,
    float* __restrict__ scores) {
  const int wave = threadIdx.x >> 5;
  const int lane = threadIdx.x & 31;
  const int tile = blockIdx.x * 4 + wave;      // 0 .. B*T/16-1
  const int b    = tile >> 9;                  // tile / (T/16)
  const int t0   = (tile & 511) << 4;          // 16-row tile origin
  const int m    = lane & 15;
  const int koff = (lane >> 4) << 1;           // 0 for lanes 0-15, 2 for 16-31

  // Wave-uniform reduction of the Dense kernel/bias (scalar loads).
  float wsum = 0.f, bsum = 0.f;
  for (int u = 0; u < U_; ++u) { wsum += W[u]; bsum += bias[u]; }

  const float* __restrict__ qb  = query  + (size_t)b * D_;
  const float* __restrict__ row = values + ((size_t)(b * T_ + t0 + m)) * D_;

  v8f acc = {0.f, 0.f, 0.f, 0.f, 0.f, 0.f, 0.f, 0.f};
  for (int d0 = 0; d0 < D_; d0 += 4) {
    v2f a  = *(const v2f*)(row + d0 + koff);   // A[m, k..k+1]
    v2f bq = *(const v2f*)(qb  + d0 + koff);   // B[k..k+1, n] = q[k] (repl.)
    acc = __builtin_amdgcn_wmma_f32_16x16x4_f32(
        /*neg_a=*/false, a, /*neg_b=*/false, bq,
        /*c_mod=*/(short)0, acc, /*reuse_a=*/false, /*reuse_b=*/false);
  }

  if (lane == 0 || lane == 16) {
    const int roff = (lane >> 4) << 3;         // rows 0-7 or 8-15
    float* out = scores + (size_t)b * T_ + t0 + roff;
#pragma unroll
    for (int r = 0; r < 8; ++r) out[r] = acc[r] * wsum + bsum;
  }
}

// ---------------------------------------------------------------------------
// Kernel 2: per-batch softmax statistics (max, 1/sum-exp). One block / batch.
// ---------------------------------------------------------------------------
__global__ __launch_bounds__(256) void softmax_stats(
    const float* __restrict__ scores, float* __restrict__ stats) {
  __shared__ float red[256];
  const int b   = blockIdx.x;
  const int tid = threadIdx.x;
  const float* s = scores + (size_t)b * T_;

  float m = -INFINITY;
  for (int t = tid; t < T_; t += 256) m = fmaxf(m, s[t]);
  red[tid] = m; __syncthreads();
  for (int k = 128; k > 0; k >>= 1) {
    if (tid < k) red[tid] = fmaxf(red[tid], red[tid + k]);
    __syncthreads();
  }
  const float mx = red[0];
  __syncthreads();

  float sum = 0.f;
  for (int t = tid; t < T_; t += 256) sum += expf(s[t] - mx);
  red[tid] = sum; __syncthreads();
  for (int k = 128; k > 0; k >>= 1) {
    if (tid < k) red[tid] += red[tid + k];
    __syncthreads();
  }
  if (tid == 0) { stats[b] = mx; stats[16 + b] = 1.0f / red[0]; }
}

// ---------------------------------------------------------------------------
// Kernel 3: attn[b,t] = exp(score-max)*invsum (stored to d_out) + per-block
// partial context over a 256-row t-chunk. blockIdx.x = b*32 + chunk.
// values rows are read fully coalesced (thread j -> columns 2j, 2j+1).
// ---------------------------------------------------------------------------
__global__ __launch_bounds__(256) void attn_ctx(
    const float* __restrict__ values,
    const float* __restrict__ scores,
    const float* __restrict__ stats,
    float* __restrict__ attn_out,
    float* __restrict__ partials) {
  __shared__ float a_s[256];
  const int tid   = threadIdx.x;
  const int b     = blockIdx.x >> 5;
  const int chunk = blockIdx.x & 31;
  const int t0    = chunk * 256;

  const float mx  = stats[b];
  const float inv = stats[16 + b];

  const int t = t0 + tid;
  const float a = expf(scores[(size_t)b * T_ + t] - mx) * inv;
  attn_out[(size_t)b * T_ + t] = a;
  a_s[tid] = a;
  __syncthreads();

  const int d = tid * 2;
  float acc0 = 0.f, acc1 = 0.f;
  const float* base = values + ((size_t)(b * T_ + t0)) * D_;
  for (int tl = 0; tl < 256; ++tl) {
    const float w = a_s[tl];
    const float* rowp = base + (size_t)tl * D_;
    acc0 += w * rowp[d];
    acc1 += w * rowp[d + 1];
  }
  float* p = partials + (size_t)blockIdx.x * D_ + d;
  p[0] = acc0;
  p[1] = acc1;
}

// ---------------------------------------------------------------------------
// Kernel 4: deterministic 32-way reduction of context partials -> d_out.
// ---------------------------------------------------------------------------
__global__ __launch_bounds__(256) void ctx_reduce(
    const float* __restrict__ partials, float* __restrict__ ctx_out) {
  const int idx = blockIdx.x * 256 + threadIdx.x;  // 0 .. B*D-1
  const int b = idx >> 9;
  const int d = idx & 511;
  const float* p = partials + ((size_t)b * 32) * D_ + d;
  float s = 0.f;
#pragma unroll
  for (int c = 0; c < 32; ++c) s += p[(size_t)c * D_];
  ctx_out[idx] = s;
}

extern "C" void kernel_launch(void* const* d_in, const int* in_sizes, int n_in,
                              void* d_out, int out_size, void* d_ws, size_t ws_size,
                              hipStream_t stream) {
  const float* query  = (const float*)d_in[0];  // [B, D]
  const float* values = (const float*)d_in[1];  // [B, T, D]
  const float* W      = (const float*)d_in[2];  // [1, U]
  const float* bias   = (const float*)d_in[3];  // [U]

  float* out      = (float*)d_out;
  float* ctx_out  = out;            // context_vector [B, D]
  float* attn_out = out + B_ * D_;  // attention_weights [B, T, 1]

  // Workspace layout (floats): scores[B*T] | stats[32] | partials[B*32*D]
  float* scores   = (float*)d_ws;
  float* stats    = scores + (size_t)B_ * T_;
  float* partials = stats + 32;

  scores_wmma<<<(B_ * T_ / 16) / 4, 128, 0, stream>>>(query, values, W, bias, scores);
  softmax_stats<<<B_, 256, 0, stream>>>(scores, stats);
  attn_ctx<<<B_ * 32, 256, 0, stream>>>(values, scores, stats, attn_out, partials);
  ctx_reduce<<<(B_ * D_) / 256, 256, 0, stream>>>(partials, ctx_out);
}